// KGAttentionLayer_50775103373665
// MI455X (gfx1250) — compile-verified
//
#include <hip/hip_runtime.h>
#include <hip/hip_bf16.h>

// ---------------------------------------------------------------------------
// KG attention layer, factorized:
//   score_e = leaky_relu( (v_e * i_e) . Rp[rel_e]  +  rb[rel_e] )
//   Rp = relation_emb @ (W1 + W2)   (computed once via V_WMMA_F32_16X16X4_F32)
//   rb = relation_emb @ b
// then segment-softmax over sorted edge_item and weighted entity aggregation.
// ---------------------------------------------------------------------------

typedef float v2f __attribute__((ext_vector_type(2)));
typedef float v8f __attribute__((ext_vector_type(8)));

#define DIM 64

// ---------------------------------------------------------------------------
// Kernel 1: Rp[64,64] = relation_emb[64,64] @ W_sum[64,64]  via f32 WMMA.
//   W_sum[j][n] = W[j*128 + n] + W[j*128 + 64 + n]
//   rb[r] = sum_j relation_emb[r][j] * b[j]
// One block, 128 threads = 4 waves. Wave w computes rows [16w,16w+16), all 64
// cols as 4 tiles, K=64 reduction in 16 steps of 4 (V_WMMA_F32_16X16X4_F32).
// EXEC is all-ones for every wave (uniform control flow around WMMA).
// ---------------------------------------------------------------------------
__global__ __launch_bounds__(128)
void kg_precompute_kernel(const float* __restrict__ rel_emb, // [64,64]
                          const float* __restrict__ W,       // [64,128]
                          const float* __restrict__ b,       // [64]
                          float* __restrict__ Rp,            // [64,64]
                          float* __restrict__ rb)            // [64]
{
    const int tid  = threadIdx.x;     // 0..127
    const int wave = tid >> 5;        // 0..3 -> row block
    const int lane = tid & 31;
    const int lh   = lane >> 4;       // 0/1 (K-half select, ISA 16x4 A layout)
    const int l15  = lane & 15;

    v8f acc0 = {}, acc1 = {}, acc2 = {}, acc3 = {};

    const int am = wave * 16 + l15;   // A-matrix row (M = relation index)

    for (int kb = 0; kb < 16; ++kb) {
        const int k0 = kb * 4 + lh * 2;           // K index for component .x
        // A fragment: 16x4 f32 (M striped over lanes 0..15 / 16..31)
        v2f a;
        a.x = rel_emb[am * DIM + k0];
        a.y = rel_emb[am * DIM + k0 + 1];

        // B fragments: 4x16 f32, N = 16*t + l15, K = k0 + v
        const int row0 = k0 * 128;
        const int row1 = (k0 + 1) * 128;
        {
            const int n = l15;
            v2f bb; bb.x = W[row0 + n] + W[row0 + 64 + n];
                    bb.y = W[row1 + n] + W[row1 + 64 + n];
            acc0 = __builtin_amdgcn_wmma_f32_16x16x4_f32(false, a, false, bb,
                                                         (short)0, acc0, false, false);
        }
        {
            const int n = 16 + l15;
            v2f bb; bb.x = W[row0 + n] + W[row0 + 64 + n];
                    bb.y = W[row1 + n] + W[row1 + 64 + n];
            acc1 = __builtin_amdgcn_wmma_f32_16x16x4_f32(false, a, false, bb,
                                                         (short)0, acc1, false, false);
        }
        {
            const int n = 32 + l15;
            v2f bb; bb.x = W[row0 + n] + W[row0 + 64 + n];
                    bb.y = W[row1 + n] + W[row1 + 64 + n];
            acc2 = __builtin_amdgcn_wmma_f32_16x16x4_f32(false, a, false, bb,
                                                         (short)0, acc2, false, false);
        }
        {
            const int n = 48 + l15;
            v2f bb; bb.x = W[row0 + n] + W[row0 + 64 + n];
                    bb.y = W[row1 + n] + W[row1 + 64 + n];
            acc3 = __builtin_amdgcn_wmma_f32_16x16x4_f32(false, a, false, bb,
                                                         (short)0, acc3, false, false);
        }
    }

    // D layout: VGPR p -> row = 16*wave + p + 8*lh, col = 16*t + l15
    #pragma unroll
    for (int p = 0; p < 8; ++p) {
        const int m = wave * 16 + p + lh * 8;
        Rp[m * DIM +      l15] = acc0[p];
        Rp[m * DIM + 16 + l15] = acc1[p];
        Rp[m * DIM + 32 + l15] = acc2[p];
        Rp[m * DIM + 48 + l15] = acc3[p];
    }

    if (tid < 64) {
        float s = 0.f;
        #pragma unroll 8
        for (int j = 0; j < DIM; ++j) s += rel_emb[tid * DIM + j] * b[j];
        rb[tid] = s;
    }
}

// ---------------------------------------------------------------------------
// Kernel 2: per-edge score. 8 lanes per edge, each lane covers 8 dims
// (2x float4 loads per operand -> one 256B coalesced row per 8-lane group).
// ---------------------------------------------------------------------------
__global__ __launch_bounds__(256)
void kg_score_kernel(const float* __restrict__ item_emb,
                     const float* __restrict__ ent_emb,
                     const int*   __restrict__ e_item,
                     const int*   __restrict__ e_rel,
                     const int*   __restrict__ e_ent,
                     const float* __restrict__ Rp,
                     const float* __restrict__ rb,
                     float* __restrict__ score,
                     int E)
{
    const int e = (blockIdx.x * blockDim.x + threadIdx.x) >> 3; // edge id
    const int l = threadIdx.x & 7;                              // sub-lane
    if (e >= E) return;

    const int it = e_item[e];
    const int rl = e_rel[e];
    const int en = e_ent[e];

    const float4* ip = (const float4*)(item_emb + (size_t)it * DIM + l * 8);
    const float4* vp = (const float4*)(ent_emb  + (size_t)en * DIM + l * 8);
    const float4* rp = (const float4*)(Rp       + (size_t)rl * DIM + l * 8);

    const float4 a0 = ip[0], a1 = ip[1];
    const float4 b0 = vp[0], b1 = vp[1];
    const float4 c0 = rp[0], c1 = rp[1];

    float p = a0.x * b0.x * c0.x + a0.y * b0.y * c0.y
            + a0.z * b0.z * c0.z + a0.w * b0.w * c0.w
            + a1.x * b1.x * c1.x + a1.y * b1.y * c1.y
            + a1.z * b1.z * c1.z + a1.w * b1.w * c1.w;

    p += __shfl_xor(p, 1, 8);
    p += __shfl_xor(p, 2, 8);
    p += __shfl_xor(p, 4, 8);

    if (l == 0) {
        float s = p + rb[rl];
        score[e] = (s > 0.f) ? s : 0.2f * s;   // leaky_relu(0.2)
    }
}

// ---------------------------------------------------------------------------
// Kernel 3: per-item segment softmax + aggregation. One wave32 per item;
// binary search the sorted edge_item for [start,end), wave-reduce max and
// exp-sum, then accumulate alpha-weighted v_vec rows (2 floats per lane).
// ---------------------------------------------------------------------------
__global__ __launch_bounds__(256)
void kg_aggregate_kernel(const float* __restrict__ item_emb,
                         const float* __restrict__ ent_emb,
                         const int*   __restrict__ e_item,
                         const int*   __restrict__ e_ent,
                         const float* __restrict__ score,
                         float* __restrict__ out,
                         int E, int nItems)
{
    const int item = (blockIdx.x * blockDim.x + threadIdx.x) >> 5;
    const int lane = threadIdx.x & 31;
    if (item >= nItems) return;

    // lower_bound(item) and lower_bound(item+1) — same path for all lanes
    int lo = 0, hi = E;
    while (lo < hi) { int mid = (lo + hi) >> 1; if (e_item[mid] < item) lo = mid + 1; else hi = mid; }
    const int start = lo;
    hi = E;
    while (lo < hi) { int mid = (lo + hi) >> 1; if (e_item[mid] <= item) lo = mid + 1; else hi = mid; }
    const int end = lo;

    float* orow = out + (size_t)item * DIM + lane * 2;

    if (start == end) {
        // no neighbors: pass item embedding through
        const float2 s = *(const float2*)(item_emb + (size_t)item * DIM + lane * 2);
        *(float2*)orow = s;
        return;
    }

    // segment max
    float m = -3.402823466e38f;
    for (int e = start + lane; e < end; e += 32) m = fmaxf(m, score[e]);
    #pragma unroll
    for (int off = 16; off > 0; off >>= 1) m = fmaxf(m, __shfl_xor(m, off, 32));

    // segment sum of exp
    float den = 0.f;
    for (int e = start + lane; e < end; e += 32) den += __expf(score[e] - m);
    #pragma unroll
    for (int off = 16; off > 0; off >>= 1) den += __shfl_xor(den, off, 32);

    // weighted aggregation of entity rows (sequential over edges, 2 dims/lane)
    float a0 = 0.f, a1 = 0.f;
    for (int e = start; e < end; ++e) {
        const float w  = __expf(score[e] - m);     // broadcast load per wave
        const int  en  = e_ent[e];                 // broadcast load per wave
        const float2 v = *(const float2*)(ent_emb + (size_t)en * DIM + lane * 2);
        a0 += w * v.x;
        a1 += w * v.y;
    }
    const float inv = 1.0f / den;
    float2 r; r.x = a0 * inv; r.y = a1 * inv;
    *(float2*)orow = r;
}

// ---------------------------------------------------------------------------
// Launcher. Input order (setup_inputs): item_emb, entity_emb, relation_emb,
// W_weight, W_bias, edge_item, edge_rel, edge_ent. Integers arrive as int32.
// Workspace: Rp[4096] f32 | rb[64] f32 | score[E] f32  (~4.2 MB).
// ---------------------------------------------------------------------------
extern "C" void kernel_launch(void* const* d_in, const int* in_sizes, int n_in,
                              void* d_out, int out_size, void* d_ws, size_t ws_size,
                              hipStream_t stream)
{
    const float* item_emb = (const float*)d_in[0];
    const float* ent_emb  = (const float*)d_in[1];
    const float* rel_emb  = (const float*)d_in[2];
    const float* W        = (const float*)d_in[3];
    const float* bias     = (const float*)d_in[4];
    const int*   e_item   = (const int*)d_in[5];
    const int*   e_rel    = (const int*)d_in[6];
    const int*   e_ent    = (const int*)d_in[7];

    const int E      = in_sizes[5];
    const int nItems = in_sizes[0] / DIM;

    float* Rp    = (float*)d_ws;
    float* rb    = Rp + 64 * DIM;
    float* score = rb + DIM;

    kg_precompute_kernel<<<1, 128, 0, stream>>>(rel_emb, W, bias, Rp, rb);

    {
        const long long threads = (long long)E * 8;
        const int blocks = (int)((threads + 255) / 256);
        kg_score_kernel<<<blocks, 256, 0, stream>>>(item_emb, ent_emb,
                                                    e_item, e_rel, e_ent,
                                                    Rp, rb, score, E);
    }
    {
        const long long threads = (long long)nItems * 32;
        const int blocks = (int)((threads + 255) / 256);
        kg_aggregate_kernel<<<blocks, 256, 0, stream>>>(item_emb, ent_emb,
                                                        e_item, e_ent,
                                                        score, (float*)d_out,
                                                        E, nItems);
    }
}